// GCN_2D_12352325943369
// MI455X (gfx1250) — compile-verified
//
#include <hip/hip_runtime.h>

#define N_NODES  100000
#define N_EDGES  800000
#define IN_DIM   64
#define HID      128
#define N_GRAPHS 256
#define LN_EPS   1e-5f

typedef __attribute__((ext_vector_type(2))) float v2f;
typedef __attribute__((ext_vector_type(8))) float v8f;

// ---------------- degree / normalization ----------------
__global__ void deg_init_kernel(float* deg, int n) {
    int i = blockIdx.x * blockDim.x + threadIdx.x;
    if (i < n) deg[i] = 1.0f;                       // implicit self-loop
}

__global__ void deg_accum_kernel(const int* __restrict__ dst, float* deg, int e) {
    int i = blockIdx.x * blockDim.x + threadIdx.x;
    if (i < e) atomicAdd(&deg[dst[i]], 1.0f);
}

__global__ void deg_rsqrt_kernel(float* deg, int n) {
    int i = blockIdx.x * blockDim.x + threadIdx.x;
    if (i < n) deg[i] = rsqrtf(deg[i]);             // deg >= 1
}

// ---------------- dense GEMM via fp32 WMMA ----------------
// C[M,128] = A[M,K] @ B[K,128], row-major. One block = 16 rows x 128 cols,
// 8 waves, each wave one 16x16 tile via V_WMMA_F32_16X16X4_F32.
// M must be a multiple of 16 (100000 = 6250*16).
template<int K>
__global__ __launch_bounds__(256) void gemm_wmma_kernel(
    const float* __restrict__ A, const float* __restrict__ B,
    float* __restrict__ C)
{
    const int lane = threadIdx.x & 31;
    const int wave = threadIdx.x >> 5;          // 0..7 -> N tile
    const int l15  = lane & 15;
    const int hi   = lane >> 4;                 // 0 or 1 (half-wave)
    const int row  = blockIdx.x * 16 + l15;     // A row for this lane
    const int col  = wave * 16 + l15;           // B/C column
    const int kh   = hi * 2;                    // K sub-offset per half-wave

    const float* __restrict__ Arow = A + (size_t)row * K;
    v8f acc = {};
    #pragma unroll
    for (int k0 = 0; k0 < K; k0 += 4) {
        v2f a, b;
        // A 16x4 fragment: VGPR0 = K {0|2}, VGPR1 = K {1|3} per half-wave
        a.x = Arow[k0 + kh + 0];
        a.y = Arow[k0 + kh + 1];
        // B 4x16 fragment: same K striping, column across lanes
        b.x = B[(size_t)(k0 + kh + 0) * HID + col];
        b.y = B[(size_t)(k0 + kh + 1) * HID + col];
        acc = __builtin_amdgcn_wmma_f32_16x16x4_f32(
            /*neg_a=*/false, a, /*neg_b=*/false, b,
            /*c_mod=*/(short)0, acc, /*reuse_a=*/false, /*reuse_b=*/false);
    }
    // C/D layout: VGPR r -> row r (lanes 0-15) / row r+8 (lanes 16-31)
    const int rbase = blockIdx.x * 16 + hi * 8;
    #pragma unroll
    for (int r = 0; r < 8; ++r)
        C[(size_t)(rbase + r) * HID + col] = acc[r];
}

// ---------------- edge scatter (message passing) ----------------
// One wave per edge; 32 lanes x float4 = 128 features, 512B coalesced
// gather from h[src], fp32 atomic scatter into agg[dst] (L2-resident).
__global__ __launch_bounds__(256) void edge_scatter_kernel(
    const int* __restrict__ src, const int* __restrict__ dst,
    const float* __restrict__ dinv, const float* __restrict__ h,
    float* __restrict__ agg, int e)
{
    const int tid  = blockIdx.x * blockDim.x + threadIdx.x;
    const int wid  = tid >> 5;
    const int lane = tid & 31;
    if (wid >= e) return;
    const int s = src[wid];
    const int d = dst[wid];
    const float norm = dinv[s] * dinv[d];
    const float4 v = ((const float4*)(h + (size_t)s * HID))[lane];
    float* ad = agg + (size_t)d * HID + lane * 4;
    atomicAdd(ad + 0, v.x * norm);
    atomicAdd(ad + 1, v.y * norm);
    atomicAdd(ad + 2, v.z * norm);
    atomicAdd(ad + 3, v.w * norm);
}

// agg = relu(agg + h*dinv^2 + bias)
__global__ __launch_bounds__(256) void selfloop_bias_relu_kernel(
    float* __restrict__ agg, const float* __restrict__ h,
    const float* __restrict__ dinv, const float* __restrict__ bias, int n)
{
    const int gid = blockIdx.x * blockDim.x + threadIdx.x;
    if (gid >= n * HID) return;
    const int node = gid >> 7;
    const int f    = gid & (HID - 1);
    const float di = dinv[node];
    const float v  = agg[gid] + h[gid] * di * di + bias[f];
    agg[gid] = fmaxf(v, 0.0f);
}

// ---------------- global mean pool ----------------
__global__ __launch_bounds__(256) void pool_accum_kernel(
    const float* __restrict__ h, const int* __restrict__ batch,
    float* __restrict__ sums, float* __restrict__ cnt, int n)
{
    const int tid  = blockIdx.x * blockDim.x + threadIdx.x;
    const int wid  = tid >> 5;
    const int lane = tid & 31;
    if (wid >= n) return;
    const int g = batch[wid];
    const float4 v = ((const float4*)(h + (size_t)wid * HID))[lane];
    float* sp = sums + (size_t)g * HID + lane * 4;
    atomicAdd(sp + 0, v.x);
    atomicAdd(sp + 1, v.y);
    atomicAdd(sp + 2, v.z);
    atomicAdd(sp + 3, v.w);
    if (lane == 0) atomicAdd(&cnt[g], 1.0f);
}

// ---------------- FC head + LayerNorm (one block per graph) ----------------
__global__ __launch_bounds__(HID) void fc_ln_kernel(
    const float* __restrict__ sums, const float* __restrict__ cnt,
    const float* __restrict__ Wfc, const float* __restrict__ bfc,
    const float* __restrict__ gamma, const float* __restrict__ beta,
    float* __restrict__ out)
{
    __shared__ float pooled[HID];
    __shared__ float r1[HID];
    __shared__ float r2[HID];
    const int g = blockIdx.x;
    const int f = threadIdx.x;

    const float c = fmaxf(cnt[g], 1.0f);
    pooled[f] = sums[(size_t)g * HID + f] / c;
    __syncthreads();

    float y = bfc[f];
    #pragma unroll 8
    for (int k = 0; k < HID; ++k)
        y = fmaf(pooled[k], Wfc[(size_t)k * HID + f], y);

    r1[f] = y;
    r2[f] = y * y;
    __syncthreads();
    for (int s = HID / 2; s > 0; s >>= 1) {
        if (f < s) { r1[f] += r1[f + s]; r2[f] += r2[f + s]; }
        __syncthreads();
    }
    const float mu  = r1[0] * (1.0f / HID);
    const float var = r2[0] * (1.0f / HID) - mu * mu;
    out[(size_t)g * HID + f] = (y - mu) * rsqrtf(var + LN_EPS) * gamma[f] + beta[f];
}

// ---------------- host-side launch ----------------
extern "C" void kernel_launch(void* const* d_in, const int* in_sizes, int n_in,
                              void* d_out, int out_size, void* d_ws, size_t ws_size,
                              hipStream_t stream)
{
    const float* x     = (const float*)d_in[0];
    const int*   eidx  = (const int*)  d_in[1];   // [2, E]
    const int*   batch = (const int*)  d_in[2];
    const float* W1    = (const float*)d_in[3];
    const float* b1    = (const float*)d_in[4];
    const float* W2    = (const float*)d_in[5];
    const float* b2    = (const float*)d_in[6];
    const float* Wfc   = (const float*)d_in[7];
    const float* bfc   = (const float*)d_in[8];
    const float* gamma = (const float*)d_in[9];
    const float* beta  = (const float*)d_in[10];

    const int* src = eidx;
    const int* dst = eidx + N_EDGES;

    // workspace layout (floats), 256B-aligned regions
    float* dinv = (float*)d_ws;                          // N_NODES (as deg, then dinv)
    float* bufA = dinv + 100352;                         // N_NODES*HID (h1 / h2)
    float* bufB = bufA + (size_t)N_NODES * HID;          // N_NODES*HID (agg / relu)
    float* sums = bufB + (size_t)N_NODES * HID;          // N_GRAPHS*HID
    float* cnt  = sums + (size_t)N_GRAPHS * HID;         // N_GRAPHS

    const size_t featBytes = (size_t)N_NODES * HID * sizeof(float);

    // degrees -> dinv
    deg_init_kernel <<<(N_NODES + 255) / 256, 256, 0, stream>>>(dinv, N_NODES);
    deg_accum_kernel<<<(N_EDGES + 255) / 256, 256, 0, stream>>>(dst, dinv, N_EDGES);
    deg_rsqrt_kernel<<<(N_NODES + 255) / 256, 256, 0, stream>>>(dinv, N_NODES);

    // ---- layer 1: h1 = x @ W1 ; aggregate ; relu ----
    gemm_wmma_kernel<IN_DIM><<<N_NODES / 16, 256, 0, stream>>>(x, W1, bufA);
    hipMemsetAsync(bufB, 0, featBytes, stream);
    edge_scatter_kernel<<<(N_EDGES * 32) / 256, 256, 0, stream>>>(src, dst, dinv, bufA, bufB, N_EDGES);
    selfloop_bias_relu_kernel<<<((size_t)N_NODES * HID) / 256, 256, 0, stream>>>(bufB, bufA, dinv, b1, N_NODES);

    // ---- layer 2: h2 = relu1 @ W2 ; aggregate ; relu ----
    gemm_wmma_kernel<HID><<<N_NODES / 16, 256, 0, stream>>>(bufB, W2, bufA);
    hipMemsetAsync(bufB, 0, featBytes, stream);
    edge_scatter_kernel<<<(N_EDGES * 32) / 256, 256, 0, stream>>>(src, dst, dinv, bufA, bufB, N_EDGES);
    selfloop_bias_relu_kernel<<<((size_t)N_NODES * HID) / 256, 256, 0, stream>>>(bufB, bufA, dinv, b2, N_NODES);

    // ---- pool + fc + layernorm ----
    hipMemsetAsync(sums, 0, (size_t)(N_GRAPHS * HID + N_GRAPHS) * sizeof(float), stream);
    pool_accum_kernel<<<(N_NODES * 32 + 255) / 256, 256, 0, stream>>>(bufB, batch, sums, cnt, N_NODES);
    fc_ln_kernel<<<N_GRAPHS, HID, 0, stream>>>(sums, cnt, Wfc, bfc, gamma, beta, (float*)d_out);
}